// MACELayer_40922448396308
// MI455X (gfx1250) — compile-verified
//
#include <hip/hip_runtime.h>

#define F 64
#define INV_SQRT_AVG 0.25f   // 1/sqrt(16)
#define LDSB_STRIDE 66       // padded column stride for transposed B in LDS

typedef __attribute__((ext_vector_type(2))) float v2f;
typedef __attribute__((ext_vector_type(8))) float v8f;

// Stage a 64x64 row-major B into LDS, transposed with padding:
// ldsB[n*LDSB_STRIDE + k] = B[k*64 + n]  -> (k,k+1) pairs contiguous, 8B aligned.
__device__ __forceinline__ void stage_B(float* ldsB, const float* __restrict__ B)
{
  for (int i = threadIdx.x; i < F * F; i += blockDim.x) {
    const int k = i >> 6;
    const int n = i & 63;
    ldsB[n * LDSB_STRIDE + k] = B[i];
  }
}

// One wave computes a 16(M) x 64(N) fp32 GEMM tile: C = A(16x64) * B(64x64).
// A addressed as A[m*ars + k*acs] (global), B pre-staged in LDS (transposed+padded).
// A slice is preloaded into registers (one load clause), then 64 back-to-back
// V_WMMA_F32_16X16X4_F32 issue from registers + ds_load_b64.
__device__ __forceinline__ void gemm16x64(const float* __restrict__ A, int ars, int acs,
                                          const float* ldsB, v8f acc[4], int lane)
{
  const int half = lane >> 4;   // 0: lanes 0-15, 1: lanes 16-31
  const int idx  = lane & 15;

  // Preload this lane's A elements: K in {4s+2*half, 4s+2*half+1}, s=0..15
  float a0[16], a1[16];
  #pragma unroll
  for (int s = 0; s < 16; ++s) {
    const int kk = s * 4 + 2 * half;
    a0[s] = A[idx * ars + kk * acs];
    a1[s] = A[idx * ars + (kk + 1) * acs];
  }

  #pragma unroll
  for (int s = 0; s < 16; ++s) {
    const int kk = s * 4 + 2 * half;
    v2f a;
    a.x = a0[s];
    a.y = a1[s];
    #pragma unroll
    for (int t = 0; t < 4; ++t) {
      const int n66 = (t * 16 + idx) * LDSB_STRIDE;
      v2f b;
      b.x = ldsB[n66 + kk];       // contiguous pair -> ds_load_b64
      b.y = ldsB[n66 + kk + 1];
      acc[t] = __builtin_amdgcn_wmma_f32_16x16x4_f32(
          /*neg_a=*/false, a, /*neg_b=*/false, b,
          /*c_mod=*/(short)0, acc[t], /*reuse_a=*/false, /*reuse_b=*/false);
    }
  }
}

// Store 16x64 tile: out[m*rs + (t*16+n)*cs]
__device__ __forceinline__ void store16x64(float* __restrict__ out, int rs, int cs,
                                           const v8f acc[4], int lane)
{
  const int half = lane >> 4;
  const int n    = lane & 15;
  #pragma unroll
  for (int t = 0; t < 4; ++t) {
    #pragma unroll
    for (int v = 0; v < 8; ++v) {
      const int m = v + 8 * half;          // ISA C/D layout: VGPR v -> rows v, v+8
      out[m * rs + (t * 16 + n) * cs] = acc[t][v];
    }
  }
}

// Kernel A: h0 = x0 @ up_W0 ; h1[:,:,c] = x1[:,:,c] @ up_W1
__global__ void up_proj_kernel(const float* __restrict__ x0, const float* __restrict__ x1,
                               const float* __restrict__ uW0, const float* __restrict__ uW1,
                               float* __restrict__ h0, float* __restrict__ h1, int nTiles)
{
  __shared__ float sB0[F * LDSB_STRIDE];
  __shared__ float sB1[F * LDSB_STRIDE];
  stage_B(sB0, uW0);
  stage_B(sB1, uW1);
  __syncthreads();

  const int lane = threadIdx.x & 31;
  const int tile = blockIdx.x * (blockDim.x >> 5) + (threadIdx.x >> 5);
  if (tile >= nTiles) return;
  const int n0 = tile * 16;

  {
    v8f acc[4] = {};
    gemm16x64(x0 + (size_t)n0 * F, F, 1, sB0, acc, lane);
    store16x64(h0 + (size_t)n0 * F, F, 1, acc, lane);
  }
  #pragma unroll
  for (int c = 0; c < 3; ++c) {
    v8f acc[4] = {};
    gemm16x64(x1 + (size_t)n0 * F * 3 + c, F * 3, 3, sB1, acc, lane);
    store16x64(h1 + (size_t)n0 * F * 3 + c, F * 3, 3, acc, lane);
  }
}

// Kernel B: edge message + segment_sum via atomics. 64 threads per edge (f = feature).
__global__ void edge_kernel(const float* __restrict__ h0, const float* __restrict__ h1,
                            const int* __restrict__ senders, const int* __restrict__ receivers,
                            const float* __restrict__ ea0, const float* __restrict__ ea1,
                            const float* __restrict__ tpw,
                            float* __restrict__ agg0, float* __restrict__ agg1, int E)
{
  __shared__ float s_tpw[4 * F];
  if (threadIdx.x < 4 * F) s_tpw[threadIdx.x] = tpw[threadIdx.x];
  __syncthreads();

  const int gid = blockIdx.x * blockDim.x + threadIdx.x;
  const int e = gid >> 6;
  const int f = gid & 63;
  if (e >= E) return;

  const int s = senders[e];
  const int r = receivers[e];
  const float a0 = ea0[e];
  const float y0 = ea1[e * 3 + 0];
  const float y1 = ea1[e * 3 + 1];
  const float y2 = ea1[e * 3 + 2];

  const float g0 = h0[(size_t)s * F + f];
  const float* gp = h1 + ((size_t)s * F + f) * 3;
  const float g1a = gp[0], g1b = gp[1], g1c = gp[2];

  const float t0 = s_tpw[f], t1 = s_tpw[F + f], t2 = s_tpw[2 * F + f], t3 = s_tpw[3 * F + f];

  const float dot = g1a * y0 + g1b * y1 + g1c * y2;
  const float m0 = t0 * g0 * a0 + t1 * dot;
  const float ta = t2 * g0;
  const float tb = t3 * a0;

  atomicAdd(&agg0[(size_t)r * F + f], m0);
  float* ap = agg1 + ((size_t)r * F + f) * 3;
  atomicAdd(&ap[0], ta * y0 + tb * g1a);
  atomicAdd(&ap[1], ta * y1 + tb * g1b);
  atomicAdd(&ap[2], ta * y2 + tb * g1c);
}

// Kernel C: down-projection (WMMA) + skip (per-species GEMV, VALU) + polynomial + readout
__global__ void finalize_kernel(const float* __restrict__ agg0, const float* __restrict__ agg1,
                                const float* __restrict__ dW0, const float* __restrict__ dW1,
                                const float* __restrict__ x0, const float* __restrict__ x1,
                                const int* __restrict__ spec_arr,
                                const float* __restrict__ sW0, const float* __restrict__ sW1,
                                const float* __restrict__ pb0, const float* __restrict__ pb1,
                                const float* __restrict__ rW,
                                float* __restrict__ outN, float* __restrict__ outF0,
                                float* __restrict__ outF1, int nTiles)
{
  __shared__ float sB0[F * LDSB_STRIDE];
  __shared__ float sB1[F * LDSB_STRIDE];
  stage_B(sB0, dW0);
  stage_B(sB1, dW1);
  __syncthreads();

  const int lane = threadIdx.x & 31;
  const int tile = blockIdx.x * (blockDim.x >> 5) + (threadIdx.x >> 5);
  if (tile >= nTiles) return;
  const int n0 = tile * 16;
  const int half = lane >> 4;
  const int col  = lane & 15;

  v8f d0acc[4] = {};
  gemm16x64(agg0 + (size_t)n0 * F, F, 1, sB0, d0acc, lane);
  v8f d1acc[3][4] = {};
  #pragma unroll
  for (int c = 0; c < 3; ++c)
    gemm16x64(agg1 + (size_t)n0 * F * 3 + c, F * 3, 3, sB1, d1acc[c], lane);

  float pr[8];
  #pragma unroll
  for (int v = 0; v < 8; ++v) pr[v] = 0.0f;

  #pragma unroll
  for (int t = 0; t < 4; ++t) {
    const int g = t * 16 + col;
    const float rw = rW[g];
    #pragma unroll
    for (int v = 0; v < 8; ++v) {
      const int node = n0 + v + 8 * half;
      const int spec = spec_arr[node];

      const float d0v = d0acc[t][v]    * INV_SQRT_AVG;
      const float da  = d1acc[0][t][v] * INV_SQRT_AVG;
      const float db  = d1acc[1][t][v] * INV_SQRT_AVG;
      const float dc  = d1acc[2][t][v] * INV_SQRT_AVG;

      // skip connections: sc0 = x0[node] . skip_W0[spec][:,g]; sc1 shares W1 across 3 channels
      const float* W0  = sW0 + (size_t)spec * F * F + g;
      const float* W1  = sW1 + (size_t)spec * F * F + g;
      const float* xr0 = x0 + (size_t)node * F;
      const float* xr1 = x1 + (size_t)node * F * 3;
      float s0 = 0.f, s1a = 0.f, s1b = 0.f, s1c = 0.f;
      for (int f = 0; f < F; ++f) {
        const float w0 = W0[f * F];
        const float w1 = W1[f * F];
        s0  += xr0[f]         * w0;
        s1a += xr1[f * 3 + 0] * w1;
        s1b += xr1[f * 3 + 1] * w1;
        s1c += xr1[f * 3 + 2] * w1;
      }

      const float vv = da * da + db * db + dc * dc;
      const float* p0 = pb0 + (size_t)spec * 5 * F + g;
      const float w00 = p0[0], w01 = p0[F], w02 = p0[2 * F], w03 = p0[3 * F], w04 = p0[4 * F];
      const float* p1 = pb1 + (size_t)spec * 4 * F + g;
      const float w10 = p1[0], w11 = p1[F], w12 = p1[2 * F], w13 = p1[3 * F];

      const float d0sq = d0v * d0v;
      const float out0 = w00 * d0v + w01 * d0sq + w02 * d0sq * d0v + w03 * vv + w04 * d0v * vv;
      const float k1   = w10 + w11 * d0v + w12 * d0sq + w13 * vv;

      const float f0  = out0 + s0;
      const float f1a = k1 * da + s1a;
      const float f1b = k1 * db + s1b;
      const float f1c = k1 * dc + s1c;

      outF0[(size_t)node * F + g] = f0;
      float* fp = outF1 + ((size_t)node * F + g) * 3;
      fp[0] = f1a; fp[1] = f1b; fp[2] = f1c;

      pr[v] += f0 * rw;
    }
  }

  // readout: reduce each pr[v] over the 16 lanes of this half (wave32, xor<16 stays in half)
  #pragma unroll
  for (int v = 0; v < 8; ++v) {
    float s = pr[v];
    s += __shfl_xor(s, 1, 32);
    s += __shfl_xor(s, 2, 32);
    s += __shfl_xor(s, 4, 32);
    s += __shfl_xor(s, 8, 32);
    if (col == 0) outN[n0 + v + 8 * half] = s;
  }
}

extern "C" void kernel_launch(void* const* d_in, const int* in_sizes, int n_in,
                              void* d_out, int out_size, void* d_ws, size_t ws_size,
                              hipStream_t stream)
{
  const float* x0   = (const float*)d_in[0];
  const float* x1   = (const float*)d_in[1];
  const int*   spec = (const int*)d_in[2];
  const int*   snd  = (const int*)d_in[3];
  const int*   rcv  = (const int*)d_in[4];
  const float* ea0  = (const float*)d_in[5];
  const float* ea1  = (const float*)d_in[6];
  const float* sW0  = (const float*)d_in[7];
  const float* sW1  = (const float*)d_in[8];
  const float* uW0  = (const float*)d_in[9];
  const float* uW1  = (const float*)d_in[10];
  const float* tpw  = (const float*)d_in[11];
  const float* dW0  = (const float*)d_in[12];
  const float* dW1  = (const float*)d_in[13];
  const float* pb0  = (const float*)d_in[14];
  const float* pb1  = (const float*)d_in[15];
  const float* rW   = (const float*)d_in[16];

  const int N = in_sizes[0] / F;   // 32768
  const int E = in_sizes[3];       // 524288

  // workspace layout: h0 | h1 | agg0 | agg1  (8 + 24 + 8 + 24 MB = 64 MB)
  float* h0   = (float*)d_ws;
  float* h1   = h0   + (size_t)N * F;
  float* agg0 = h1   + (size_t)N * F * 3;
  float* agg1 = agg0 + (size_t)N * F;

  // zero the accumulators (agg0+agg1 contiguous)
  hipMemsetAsync(agg0, 0, (size_t)N * F * 4 * sizeof(float), stream);

  const int nTiles = N / 16;                // 2048 wave-tiles
  dim3 blk(128);                            // 4 waves per block
  dim3 grd((nTiles + 3) / 4);

  up_proj_kernel<<<grd, blk, 0, stream>>>(x0, x1, uW0, uW1, h0, h1, nTiles);

  const long long totalE = (long long)E * F;
  edge_kernel<<<dim3((unsigned)((totalE + 255) / 256)), dim3(256), 0, stream>>>(
      h0, h1, snd, rcv, ea0, ea1, tpw, agg0, agg1, E);

  float* outN  = (float*)d_out;
  float* outF0 = outN + N;
  float* outF1 = outF0 + (size_t)N * F;
  finalize_kernel<<<grd, blk, 0, stream>>>(agg0, agg1, dW0, dW1, x0, x1, spec,
                                           sW0, sW1, pb0, pb1, rW,
                                           outN, outF0, outF1, nTiles);
}